// PyramidVisionTransformer_65180423685632
// MI455X (gfx1250) — compile-verified
//
#include <hip/hip_runtime.h>
#include <hip/hip_bf16.h>

typedef float v2f __attribute__((ext_vector_type(2)));
typedef float v8f __attribute__((ext_vector_type(8)));

#define ST  132   // per-wave scratch row stride (floats), bank-staggered
#define WST 68    // K-major weight stride (floats)

__device__ __forceinline__ void lds_fence() {
  asm volatile("s_wait_dscnt 0" ::: "memory");
}

// fast pow for x > 0 only (v_log_f32 + v_mul + v_exp_f32)
__device__ __forceinline__ float fpow(float x, float p) {
  return __expf(p * __logf(x));
}

// ---------------------------------------------------------------------------
// K0a: per-channel params: prm[c] = 1/softplus(scale_p[c]),
//      prm[64+c] = 1 + 4*sigmoid(power_p[c])
// ---------------------------------------------------------------------------
__global__ __launch_bounds__(64) void pvt_params(
    const float* __restrict__ scale_p, const float* __restrict__ power_p,
    float* __restrict__ prm)
{
  const int c = threadIdx.x;
  prm[c]      = 1.f / log1pf(expf(scale_p[c]));
  prm[64 + c] = 1.f + 4.f / (1.f + expf(-power_p[c]));
}

// ---------------------------------------------------------------------------
// K0b: transpose W_sr (64co x 4096k) -> Wt (4096k x 64co), coalesced writes
// ---------------------------------------------------------------------------
__global__ __launch_bounds__(1024) void pvt_wtrans(
    const float* __restrict__ Wsr, float* __restrict__ Wt)
{
  const int i = blockIdx.x * 1024 + threadIdx.x;   // 262144 elems
  const int co = i & 63, k = i >> 6;
  Wt[i] = Wsr[(size_t)co * 4096 + k];
}

// ---------------------------------------------------------------------------
// K1: 8x8 stride-8 conv (x -> x_sr, pre-LN).  One block = 16 pixels (one hr
// row) x 64 out-channels.  Patch staged in 32KB LDS chunks (8 ci at a time);
// each thread holds 4 pixel accumulators so each weight load is reused 4x.
// LDS reads are wave-broadcast (address independent of lane's co).
// ---------------------------------------------------------------------------
__global__ __launch_bounds__(256) void pvt_conv_sr(
    const float* __restrict__ x, const float* __restrict__ Wt,
    const float* __restrict__ bsr, float* __restrict__ xsr)
{
  __shared__ float sp[16 * 512];           // [wr][ (ci%8)*64 + yy*8+xx ]
  const int blk = blockIdx.x;              // B*16
  const int b = blk >> 4, hr = blk & 15;
  const int tid = threadIdx.x;
  const int co = tid & 63;
  const int qd = tid >> 6;                 // pixel quad 0..3
  float acc[4] = {0.f, 0.f, 0.f, 0.f};

  for (int g = 0; g < 8; ++g) {            // ci groups of 8
    __syncthreads();
    for (int i = tid; i < 16 * 512; i += 256) {
      int m = i >> 9, kk = i & 511;
      int ci = (g << 3) + (kk >> 6);
      int pp = kk & 63, yy = pp >> 3, xx = pp & 7;
      int tok = (hr * 8 + yy) * 128 + m * 8 + xx;
      sp[i] = x[((size_t)(b * 16384 + tok)) * 64 + ci];
    }
    __syncthreads();
    const float* wp = Wt + (size_t)(g * 512) * 64 + co;
    const float* p0 = &sp[(qd * 4) * 512];
    for (int k = 0; k < 512; ++k) {
      float w = wp[(size_t)k * 64];
      acc[0] += p0[k]        * w;
      acc[1] += p0[512 + k]  * w;
      acc[2] += p0[1024 + k] * w;
      acc[3] += p0[1536 + k] * w;
    }
  }
#pragma unroll
  for (int m = 0; m < 4; ++m) {
    int p = hr * 16 + qd * 4 + m;
    xsr[((size_t)(b * 256 + p)) * 64 + co] = acc[m] + bsr[co];
  }
}

// ---------------------------------------------------------------------------
// K2: LayerNorm + kv GEMM + k transforms.  One block per reduced token.
// ---------------------------------------------------------------------------
__global__ __launch_bounds__(128) void pvt_ln_kv(
    const float* __restrict__ xsr, const float* __restrict__ Wkv,
    const float* __restrict__ bkv, const float* __restrict__ lng,
    const float* __restrict__ lnb, const float* __restrict__ pos,
    const float* __restrict__ prm,
    float* __restrict__ kcat, float* __restrict__ v)
{
  __shared__ float xr[64], xn[64];
  __shared__ float mu_s, rs_s;
  const int row = blockIdx.x;              // b*256 + p
  const int p = row & 255;
  const int tid = threadIdx.x;
  if (tid < 64) xr[tid] = xsr[(size_t)row * 64 + tid];
  __syncthreads();
  if (tid == 0) {
    float s = 0.f;
    for (int i = 0; i < 64; ++i) s += xr[i];
    float mu = s * (1.f / 64.f);
    float vv = 0.f;
    for (int i = 0; i < 64; ++i) { float d = xr[i] - mu; vv += d * d; }
    mu_s = mu; rs_s = rsqrtf(vv * (1.f / 64.f) + 1e-5f);
  }
  __syncthreads();
  if (tid < 64) xn[tid] = (xr[tid] - mu_s) * rs_s * lng[tid] + lnb[tid];
  __syncthreads();
  float acc = bkv[tid];
  for (int c = 0; c < 64; ++c) acc += xn[c] * Wkv[c * 128 + tid];
  if (tid < 64) {
    float kk = (acc + pos[p * 64 + tid]) * prm[tid];
    float pw = prm[64 + tid];
    kcat[(size_t)row * 128 + tid]      = kk > 0.f ? fpow(kk, pw)  : 0.f;
    kcat[(size_t)row * 128 + 64 + tid] = kk < 0.f ? fpow(-kk, pw) : 0.f;
  } else {
    v[(size_t)row * 64 + (tid - 64)] = acc;
  }
}

// ---------------------------------------------------------------------------
// K3: per-batch 128x80 B-matrix:
//   cols  0..31 : kv_sim[d][e] = (1/256) sum_p kcat[p][d] v[p][e]
//   cols 32..63 : kv_opp[(d+64)%128][e]   (q_opp rotation folded into B)
//   col  64     : k_mean[d];  col 65: k_mean[(d+64)%128]
// ---------------------------------------------------------------------------
__global__ __launch_bounds__(1024) void pvt_kvmat(
    const float* __restrict__ kcat, const float* __restrict__ v,
    float* __restrict__ Bmat)
{
  __shared__ float km[128];
  const int b = blockIdx.x;
  const int tid = threadIdx.x;
  const float* kb = kcat + (size_t)b * 256 * 128;
  const float* vb = v + (size_t)b * 256 * 64;
  if (tid < 128) {
    float s = 0.f;
    for (int p = 0; p < 256; ++p) s += kb[p * 128 + tid];
    km[tid] = s * (1.f / 256.f);
  }
  __syncthreads();
  for (int idx = tid; idx < 128 * 80; idx += 1024) {
    int d = idx / 80, col = idx - d * 80;
    float val = 0.f;
    if (col < 64) {
      int dd = (col < 32) ? d : ((d + 64) & 127);
      int eo = col & 31;
      if (col >= 32) eo += 32;
      float s = 0.f;
      for (int p = 0; p < 256; ++p) s += kb[p * 128 + dd] * vb[p * 64 + eo];
      val = s * (1.f / 256.f);
    } else if (col == 64) val = km[d];
    else if (col == 65)   val = km[(d + 64) & 127];
    Bmat[((size_t)b * 128 + d) * 80 + col] = val;
  }
}

// ---------------------------------------------------------------------------
// K4: depthwise 5x5 conv (pad 2) on v as (B,64,16,16) -> dwc map
// ---------------------------------------------------------------------------
__global__ __launch_bounds__(256) void pvt_dwc(
    const float* __restrict__ v, const float* __restrict__ Wd,
    const float* __restrict__ bd, float* __restrict__ dwc)
{
  const int blk = blockIdx.x;
  const int b = blk >> 6, c = blk & 63;
  const int tid = threadIdx.x;
  const int yy = tid >> 4, xx = tid & 15;
  float acc = bd[c];
  const float* w = Wd + c * 25;
  for (int ky = 0; ky < 5; ++ky) {
    int sy = yy + ky - 2;
    if (sy < 0 || sy > 15) continue;
    for (int kx = 0; kx < 5; ++kx) {
      int sx = xx + kx - 2;
      if (sx < 0 || sx > 15) continue;
      acc += v[((size_t)b * 256 + sy * 16 + sx) * 64 + c] * w[ky * 5 + kx];
    }
  }
  dwc[(size_t)blk * 256 + tid] = acc;      // ((b*64+c)*16+yy)*16+xx
}

// ---------------------------------------------------------------------------
// K5: fused per-token pipeline, f32 WMMA 16x16x4. 128 threads = 4 waves,
// 16 tokens per wave, 64 tokens per workgroup.
//   GEMM1: X(16x64)@Wqg(64x128) -> q(4 tiles) + g(4 tiles)
//   elementwise -> q_sim(16x128) staged to LDS (transpose to A-layout)
//   GEMM2: q_sim @ B2(128x80) -> x_sim|x_opp|denoms
//   z / bilinear-upsampled dwc / gate -> t(16x64) staged to LDS
//   GEMM3: t @ Wproj(64x64) + b_proj -> out
// ---------------------------------------------------------------------------
__global__ __launch_bounds__(128) void pvt_main(
    const float* __restrict__ x, const float* __restrict__ Wqg,
    const float* __restrict__ bqg, const float* __restrict__ prm,
    const float* __restrict__ Wproj, const float* __restrict__ bproj,
    const float* __restrict__ Bmat, const float* __restrict__ dwc,
    float* __restrict__ out)
{
  __shared__ float sW1[128 * WST];     // Wqg  transposed: [n][k], k contiguous
  __shared__ float sW3[64 * WST];      // Wproj transposed: [n][k]
  __shared__ float sB2[80 * ST];       // Bmat transposed: [n][k], k<128
  __shared__ float sDwc[2 * 64 * 16];  // two dwc rows: [yy][c][x]
  __shared__ float sScr[4 * 16 * ST];  // per-wave A-staging (16 x <=128)
  __shared__ float sZ[4 * 32];         // per-wave z_sim/z_opp per token

  const int tid = threadIdx.x;
  const int wave = tid >> 5;
  const int lane = tid & 31;
  const int b = blockIdx.x >> 8;
  const int tok_wg = (blockIdx.x & 255) << 6;
  const int tok0 = tok_wg + (wave << 4);

  // prefetch this wave's x tile (4KB) behind the cooperative LDS staging
  {
    const float* xpf = x + (size_t)(b * 16384 + tok0) * 64;
    __builtin_prefetch(xpf + lane * 32, 0, 3);   // global_prefetch_b8, 128B/lane
  }

  // --- cooperative loads of workgroup-shared operands ---
  for (int i = tid; i < 64 * 128; i += 128) { int k = i >> 7, n = i & 127; sW1[n * WST + k] = Wqg[i]; }
  for (int i = tid; i < 64 * 64;  i += 128) { int k = i >> 6, n = i & 63;  sW3[n * WST + k] = Wproj[i]; }
  for (int i = tid; i < 128 * 80; i += 128) { int k = i / 80, n = i - k * 80; sB2[n * ST + k] = Bmat[((size_t)b * 128 + k) * 80 + n]; }

  // bilinear row pair for this WG (y constant over 64 consecutive tokens)
  const int y = tok_wg >> 7;
  const float fy = (float)y * 0.125f - 0.4375f;
  const float y0f = floorf(fy);
  const float wy = fy - y0f;
  const int iy0 = (int)y0f;
  const int ya = min(max(iy0, 0), 15), yb = min(max(iy0 + 1, 0), 15);
  for (int i = tid; i < 2 * 64 * 16; i += 128) {
    int yy = i >> 10, c = (i >> 4) & 63, xx = i & 15;
    int ys = yy ? yb : ya;
    sDwc[i] = dwc[((size_t)(b * 64 + c) * 16 + ys) * 16 + xx];
  }
  __syncthreads();

  float* scr = &sScr[wave * 16 * ST];
  float* zb  = &sZ[wave * 32];
  const int lm  = lane & 15;           // column / M within tile
  const int hi  = lane >> 4;           // half-wave select
  const int klo = hi << 1;             // A/B K sub-offset per ISA layout

  // --- stage X tile (16 tokens x 64 ch), coalesced ---
  {
    const float* xp = x + (size_t)(b * 16384 + tok0) * 64;
    for (int i = lane; i < 16 * 64; i += 32) { int m = i >> 6, k = i & 63; scr[m * ST + k] = xp[i]; }
  }
  lds_fence();

  // --- GEMM1: qg = X @ Wqg + b_qg ---
  v8f qa[4], ga[4];
#pragma unroll
  for (int nt = 0; nt < 8; ++nt) {
    const int n = nt * 16 + lm;
    const float bias = bqg[n];
    v8f acc = { bias, bias, bias, bias, bias, bias, bias, bias };
    const float* Bp = &sW1[n * WST];
#pragma unroll
    for (int k0 = 0; k0 < 64; k0 += 4) {
      v2f a  = *(const v2f*)&scr[lm * ST + k0 + klo];
      v2f bb = *(const v2f*)&Bp[k0 + klo];
      acc = __builtin_amdgcn_wmma_f32_16x16x4_f32(false, a, false, bb, (short)0, acc, false, false);
    }
    if (nt < 4) qa[nt] = acc; else ga[nt - 4] = acc;
  }
  lds_fence();   // WMMA operand ds-loads done before scr is overwritten

  // --- elementwise: q*inv_scale, safe_pow -> q_sim = [q_pos|q_neg] in scr ---
#pragma unroll
  for (int tq = 0; tq < 4; ++tq) {
    const int c = tq * 16 + lm;
    const float isc = prm[c];
    const float pw  = prm[64 + c];
#pragma unroll
    for (int j = 0; j < 8; ++j) {
      const int m = j + hi * 8;
      const float qv = (float)qa[tq][j] * isc;
      scr[m * ST + c]      = qv > 0.f ? fpow(qv, pw)  : 0.f;
      scr[m * ST + 64 + c] = qv < 0.f ? fpow(-qv, pw) : 0.f;
    }
  }
  lds_fence();

  // --- GEMM2: q_sim(16x128) @ B2(128x80) ---
  v8f ct[5];
#pragma unroll
  for (int nt = 0; nt < 5; ++nt) {
    const int n = nt * 16 + lm;
    v8f acc = { 0.f, 0.f, 0.f, 0.f, 0.f, 0.f, 0.f, 0.f };
    const float* Bp = &sB2[n * ST];
#pragma unroll
    for (int k0 = 0; k0 < 128; k0 += 4) {
      v2f a  = *(const v2f*)&scr[lm * ST + k0 + klo];
      v2f bb = *(const v2f*)&Bp[k0 + klo];
      acc = __builtin_amdgcn_wmma_f32_16x16x4_f32(false, a, false, bb, (short)0, acc, false, false);
    }
    ct[nt] = acc;
  }

  // --- normalizers: tile4 col0 -> z_sim, col1 -> z_opp ---
  if (lm < 2) {
#pragma unroll
    for (int j = 0; j < 8; ++j) {
      const int m = j + hi * 8;
      zb[lm * 16 + m] = 1.f / ((float)ct[4][j] + 1e-6f);
    }
  }
  lds_fence();

  // --- t = (attn*z + bilinear(dwc)) * g  -> scr (16x64) ---
#pragma unroll
  for (int tt = 0; tt < 4; ++tt) {
    const int c = tt * 16 + lm;
    const float* dA = &sDwc[c * 16];
    const float* dB = &sDwc[1024 + c * 16];
#pragma unroll
    for (int j = 0; j < 8; ++j) {
      const int m = j + hi * 8;
      const int px = (tok0 + m) & 127;
      const float fx = (float)px * 0.125f - 0.4375f;
      const float x0f = floorf(fx);
      const float wx = fx - x0f;
      const int ix0 = (int)x0f;
      const int xa = min(max(ix0, 0), 15), xb2 = min(max(ix0 + 1, 0), 15);
      const float v00 = dA[xa], v01 = dA[xb2], v10 = dB[xa], v11 = dB[xb2];
      const float vd = (v00 + (v01 - v00) * wx) * (1.f - wy) + (v10 + (v11 - v10) * wx) * wy;
      const float zv = zb[(tt < 2 ? 0 : 16) + m];
      scr[m * ST + c] = ((float)ct[tt][j] * zv + vd) * (float)ga[tt][j];
    }
  }
  lds_fence();

  // --- GEMM3: out = t @ Wproj + b_proj ---
#pragma unroll
  for (int nt = 0; nt < 4; ++nt) {
    const int n = nt * 16 + lm;
    const float bias = bproj[n];
    v8f acc = { bias, bias, bias, bias, bias, bias, bias, bias };
    const float* Bp = &sW3[n * WST];
#pragma unroll
    for (int k0 = 0; k0 < 64; k0 += 4) {
      v2f a  = *(const v2f*)&scr[lm * ST + k0 + klo];
      v2f bb = *(const v2f*)&Bp[k0 + klo];
      acc = __builtin_amdgcn_wmma_f32_16x16x4_f32(false, a, false, bb, (short)0, acc, false, false);
    }
    float* op = out + (size_t)(b * 16384 + tok0) * 64 + n;
#pragma unroll
    for (int j = 0; j < 8; ++j) {
      const int m = j + hi * 8;
      op[(size_t)m * 64] = (float)acc[j];
    }
  }
}

// ---------------------------------------------------------------------------
extern "C" void kernel_launch(void* const* d_in, const int* in_sizes, int n_in,
                              void* d_out, int out_size, void* d_ws, size_t ws_size,
                              hipStream_t stream)
{
  const float* x       = (const float*)d_in[0];
  // d_in[1]=H, d_in[2]=W (hardcoded 128)
  const float* Wqg     = (const float*)d_in[3];
  const float* bqg     = (const float*)d_in[4];
  const float* Wkv     = (const float*)d_in[5];
  const float* bkv     = (const float*)d_in[6];
  const float* Wsr     = (const float*)d_in[7];
  const float* bsr     = (const float*)d_in[8];
  const float* lng     = (const float*)d_in[9];
  const float* lnb     = (const float*)d_in[10];
  const float* pos     = (const float*)d_in[11];
  const float* scale_p = (const float*)d_in[12];
  const float* power_p = (const float*)d_in[13];
  const float* Wd      = (const float*)d_in[14];
  const float* bd      = (const float*)d_in[15];
  const float* Wproj   = (const float*)d_in[16];
  const float* bproj   = (const float*)d_in[17];
  float* out = (float*)d_out;

  float* ws   = (float*)d_ws;
  float* xsr  = ws;                    // 16*256*64  = 262144 f
  float* kcat = xsr + 262144;          // 16*256*128 = 524288 f
  float* vbuf = kcat + 524288;         // 16*256*64  = 262144 f
  float* Bmat = vbuf + 262144;         // 16*128*80  = 163840 f
  float* dwcb = Bmat + 163840;         // 16*64*256  = 262144 f
  float* prm  = dwcb + 262144;         // 128 f
  float* Wt   = prm + 128;             // 4096*64    = 262144 f

  pvt_params<<<1, 64, 0, stream>>>(scale_p, power_p, prm);
  pvt_wtrans<<<256, 1024, 0, stream>>>(Wsr, Wt);
  pvt_conv_sr<<<256, 256, 0, stream>>>(x, Wt, bsr, xsr);
  pvt_ln_kv<<<4096, 128, 0, stream>>>(xsr, Wkv, bkv, lng, lnb, pos, prm, kcat, vbuf);
  pvt_kvmat<<<16, 1024, 0, stream>>>(kcat, vbuf, Bmat);
  pvt_dwc<<<1024, 256, 0, stream>>>(vbuf, Wd, bd, dwcb);
  pvt_main<<<4096, 128, 0, stream>>>(x, Wqg, bqg, prm, Wproj, bproj, Bmat, dwcb, out);
}